// BoxDecoder_51247549775901
// MI455X (gfx1250) — compile-verified
//
#include <hip/hip_runtime.h>

typedef float v2f __attribute__((ext_vector_type(2)));
typedef float v8f __attribute__((ext_vector_type(8)));

#define DIM 256
#define NH  8
#define HD  32
#define NLV 4
#define NPT 4
#define LQ  900
#define BB  16
#define DFF 1024
#define M1  (LQ * BB)      // 14400 query tokens
#define LV  13294
#define MV  (LV * BB)      // 212704 memory tokens

// ---------------------------------------------------------------------------
// elementwise add (float4 vectorized)
// ---------------------------------------------------------------------------
__global__ __launch_bounds__(256) void add_kernel(const float* __restrict__ a,
                                                  const float* __restrict__ b,
                                                  float* __restrict__ o, int n4) {
  int i = blockIdx.x * 256 + threadIdx.x;
  if (i < n4) {
    float4 x = reinterpret_cast<const float4*>(a)[i];
    float4 y = reinterpret_cast<const float4*>(b)[i];
    reinterpret_cast<float4*>(o)[i] =
        make_float4(x.x + y.x, x.y + y.y, x.z + y.z, x.w + y.w);
  }
}

// ---------------------------------------------------------------------------
// GEMM:  Y[M,N] = X[M,K] @ W[N,K]^T + bias   (optional ReLU)
// fp32 WMMA 16x16x4, 64x64 block tile, 8 waves (each wave: one 16x32 tile).
// LDS pitches (36 / 68) chosen so WMMA fragment reads are bank-conflict free.
// ---------------------------------------------------------------------------
#define XP 36
#define WPP 68
__global__ __launch_bounds__(256) void gemm_kernel(
    const float* __restrict__ X, const float* __restrict__ W,
    const float* __restrict__ bias, float* __restrict__ Y,
    int M, int N, int K, int relu) {
  __shared__ float Xs[64 * XP];   // [m][k]
  __shared__ float Ws[32 * WPP];  // [k][n]  (transposed W tile)

  const int tid  = threadIdx.x;
  const int lane = tid & 31;
  const int l15  = lane & 15;
  const int hi   = lane >> 4;
  const int wave = tid >> 5;
  const int m0 = blockIdx.y * 64;
  const int n0 = blockIdx.x * 64;
  const int mw = (wave >> 1) * 16;  // wave row tile
  const int nw = (wave & 1) * 32;   // wave col tile (2x 16-wide WMMA)
  const int lm = tid >> 2;          // 0..63 loader row
  const int lj = (tid & 3) * 8;     // 0,8,16,24 loader k-offset

  v8f acc0 = {0.f, 0.f, 0.f, 0.f, 0.f, 0.f, 0.f, 0.f};
  v8f acc1 = {0.f, 0.f, 0.f, 0.f, 0.f, 0.f, 0.f, 0.f};

  for (int k0 = 0; k0 < K; k0 += 32) {
    {  // X tile: 64 rows x 32 k  (guard M tail)
      int gm = m0 + lm;
      float4 x0 = make_float4(0.f, 0.f, 0.f, 0.f), x1 = x0;
      if (gm < M) {
        const float4* src =
            reinterpret_cast<const float4*>(X + (size_t)gm * K + k0 + lj);
        x0 = src[0];
        x1 = src[1];
      }
      *reinterpret_cast<float4*>(&Xs[lm * XP + lj]) = x0;
      *reinterpret_cast<float4*>(&Xs[lm * XP + lj + 4]) = x1;
    }
    {  // W tile transposed into Ws[k][n]  (N is always a multiple of 64)
      int gn = n0 + lm;
      const float4* src =
          reinterpret_cast<const float4*>(W + (size_t)gn * K + k0 + lj);
      float4 w0 = src[0], w1 = src[1];
      Ws[(lj + 0) * WPP + lm] = w0.x;
      Ws[(lj + 1) * WPP + lm] = w0.y;
      Ws[(lj + 2) * WPP + lm] = w0.z;
      Ws[(lj + 3) * WPP + lm] = w0.w;
      Ws[(lj + 4) * WPP + lm] = w1.x;
      Ws[(lj + 5) * WPP + lm] = w1.y;
      Ws[(lj + 6) * WPP + lm] = w1.z;
      Ws[(lj + 7) * WPP + lm] = w1.w;
    }
    __syncthreads();
    // A frag: a[j] = A[l15][kk+2*hi+j]; B frag: b[j] = B[kk+2*hi+j][l15]
    const float* xrow  = &Xs[(mw + l15) * XP + 2 * hi];
    const float* wcol0 = &Ws[(2 * hi) * WPP + nw + l15];
    const float* wcol1 = wcol0 + 16;
#pragma unroll
    for (int kk = 0; kk < 32; kk += 4) {
      v2f a;  a.x  = xrow[kk];          a.y  = xrow[kk + 1];
      v2f b0; b0.x = wcol0[kk * WPP];   b0.y = wcol0[(kk + 1) * WPP];
      v2f b1; b1.x = wcol1[kk * WPP];   b1.y = wcol1[(kk + 1) * WPP];
      acc0 = __builtin_amdgcn_wmma_f32_16x16x4_f32(false, a, false, b0, 0,
                                                   acc0, false, false);
      acc1 = __builtin_amdgcn_wmma_f32_16x16x4_f32(false, a, false, b1, 0,
                                                   acc1, false, false);
    }
    __syncthreads();
  }

  // epilogue: C/D layout -> row (v + 8*hi), col l15
  int gn0 = n0 + nw + l15;
  float bv0 = bias[gn0], bv1 = bias[gn0 + 16];
#pragma unroll
  for (int v = 0; v < 8; ++v) {
    int gm = m0 + mw + v + 8 * hi;
    if (gm < M) {
      float r0 = acc0[v] + bv0;
      float r1 = acc1[v] + bv1;
      if (relu) { r0 = fmaxf(r0, 0.f); r1 = fmaxf(r1, 0.f); }
      Y[(size_t)gm * N + gn0]      = r0;
      Y[(size_t)gm * N + gn0 + 16] = r1;
    }
  }
}

// ---------------------------------------------------------------------------
// Flash self-attention. Block = (b, h, 32-query tile). K/V staged in LDS in
// 192-key chunks (pitch 33 -> conflict-free). Dot phase: lane = key.
// Accumulate phase: lane = channel (scalar per-lane accumulator, weights
// broadcast with readlane). Online softmax, wave32 butterfly reductions.
// ---------------------------------------------------------------------------
#define KCH  192
#define KPASS 5   // 5*192 = 960 >= 900
#define KPIT 33
__global__ __launch_bounds__(256) void flash_kernel(
    const float* __restrict__ QK,  // [M1, 512]  cols 0..255 = Q, 256..511 = K
    const float* __restrict__ V,   // [M1, 256]
    float* __restrict__ O) {       // [M1, 256]
  __shared__ float Ks[KCH * KPIT];
  __shared__ float Vs[KCH * KPIT];
  const int tid  = threadIdx.x;
  const int lane = tid & 31;
  const int wave = tid >> 5;
  int qb = blockIdx.x % 29;        // 29 query tiles of 32 cover 900
  int bh = blockIdx.x / 29;
  int h  = bh % NH;
  int b  = bh / NH;
  const float scale = 0.17677669529663687f;  // 1/sqrt(32)

  float m[4], s[4], acc[4];
#pragma unroll
  for (int qi = 0; qi < 4; ++qi) { m[qi] = -3.0e38f; s[qi] = 0.f; acc[qi] = 0.f; }

  for (int pass = 0; pass < KPASS; ++pass) {
    // stage K/V chunk (clamped key index -> finite data; tails masked later)
    for (int idx = tid; idx < KCH * HD; idx += 256) {
      int kk = idx >> 5, d = idx & 31;
      int kg = pass * KCH + kk;
      int kc = kg < LQ ? kg : (LQ - 1);
      size_t t = (size_t)kc * BB + b;
      Ks[kk * KPIT + d] = QK[t * 512 + 256 + h * HD + d];
      Vs[kk * KPIT + d] = V[t * 256 + h * HD + d];
    }
    __syncthreads();
#pragma unroll
    for (int qi = 0; qi < 4; ++qi) {
      int q = qb * 32 + wave + 8 * qi;
      if (q < LQ) {
        size_t tq = (size_t)q * BB + b;
        float qreg[32];
        const float4* qp =
            reinterpret_cast<const float4*>(QK + tq * 512 + h * HD);
#pragma unroll
        for (int j = 0; j < 8; ++j) {
          float4 qv = qp[j];
          qreg[4 * j + 0] = qv.x * scale; qreg[4 * j + 1] = qv.y * scale;
          qreg[4 * j + 2] = qv.z * scale; qreg[4 * j + 3] = qv.w * scale;
        }
        float mm = m[qi], ss = s[qi], aa = acc[qi];
        for (int c = 0; c < KCH / 32; ++c) {
          int kg = pass * KCH + c * 32 + lane;
          const float* kr = &Ks[(c * 32 + lane) * KPIT];
          float sc = 0.f;
#pragma unroll
          for (int d2 = 0; d2 < 32; ++d2) sc = fmaf(qreg[d2], kr[d2], sc);
          float score = (kg < LQ) ? sc : -3.0e38f;
          float cm = score;
#pragma unroll
          for (int off = 16; off > 0; off >>= 1)
            cm = fmaxf(cm, __shfl_xor(cm, off, 32));
          float mnew = fmaxf(mm, cm);
          float rsc  = __expf(mm - mnew);
          ss *= rsc; aa *= rsc; mm = mnew;
          float w = __expf(score - mm);
          ss += w;
          const float* vr = &Vs[(c * 32) * KPIT + lane];
#pragma unroll
          for (int l = 0; l < 32; ++l) {
            float wl = __shfl(w, l, 32);
            aa = fmaf(wl, vr[l * KPIT], aa);
          }
        }
        m[qi] = mm; s[qi] = ss; acc[qi] = aa;
      }
    }
    __syncthreads();
  }
#pragma unroll
  for (int qi = 0; qi < 4; ++qi) {
    int q = qb * 32 + wave + 8 * qi;
    if (q < LQ) {
      float ss = s[qi];
#pragma unroll
      for (int off = 16; off > 0; off >>= 1) ss += __shfl_xor(ss, off, 32);
      size_t tq = (size_t)q * BB + b;
      O[tq * 256 + h * HD + lane] = acc[qi] / ss;
    }
  }
}

// ---------------------------------------------------------------------------
// y = LayerNorm(res + delta) * g + b   (one wave per token, D=256)
// ---------------------------------------------------------------------------
__global__ __launch_bounds__(256) void ln_kernel(
    const float* __restrict__ R, const float* __restrict__ Dl,
    const float* __restrict__ G, const float* __restrict__ Bt,
    float* __restrict__ OUT, int M) {
  int lane = threadIdx.x & 31;
  int wave = threadIdx.x >> 5;
  int t = blockIdx.x * 8 + wave;
  if (t >= M) return;
  size_t base = (size_t)t * DIM;
  float v[8];
  float sum = 0.f;
#pragma unroll
  for (int j = 0; j < 8; ++j) {
    int d = lane + 32 * j;
    v[j] = R[base + d] + Dl[base + d];
    sum += v[j];
  }
#pragma unroll
  for (int off = 16; off > 0; off >>= 1) sum += __shfl_xor(sum, off, 32);
  float mean = sum * (1.0f / 256.0f);
  float var = 0.f;
#pragma unroll
  for (int j = 0; j < 8; ++j) { float dd = v[j] - mean; var = fmaf(dd, dd, var); }
#pragma unroll
  for (int off = 16; off > 0; off >>= 1) var += __shfl_xor(var, off, 32);
  var *= (1.0f / 256.0f);
  float rstd = rsqrtf(var + 1e-5f);
#pragma unroll
  for (int j = 0; j < 8; ++j) {
    int d = lane + 32 * j;
    OUT[base + d] = (v[j] - mean) * rstd * G[d] + Bt[d];
  }
}

// ---------------------------------------------------------------------------
// softmax over contiguous groups of 16 (attention weights, per (t,h))
// ---------------------------------------------------------------------------
__global__ __launch_bounds__(256) void softmax16_kernel(
    const float* __restrict__ in, float* __restrict__ out, int n) {
  int i = blockIdx.x * 256 + threadIdx.x;
  if (i >= n) return;
  const float* p = in + (size_t)i * 16;
  float mx = p[0];
#pragma unroll
  for (int j = 1; j < 16; ++j) mx = fmaxf(mx, p[j]);
  float e[16];
  float ssum = 0.f;
#pragma unroll
  for (int j = 0; j < 16; ++j) { e[j] = __expf(p[j] - mx); ssum += e[j]; }
  float inv = 1.0f / ssum;
  float* q = out + (size_t)i * 16;
#pragma unroll
  for (int j = 0; j < 16; ++j) q[j] = e[j] * inv;
}

// ---------------------------------------------------------------------------
// MS-deformable sampling: one wave per (token t, head h); lane = channel.
// Each bilinear corner is one coalesced 128B gather from VAL.
// ---------------------------------------------------------------------------
__global__ __launch_bounds__(256) void deform_kernel(
    const float* __restrict__ REF,  // (LQ,B,NL,2)
    const float* __restrict__ OFF,  // [t, 256]
    const float* __restrict__ AW,   // [t, 128] (softmaxed)
    const float* __restrict__ VAL,  // [LV*B, 256]
    float* __restrict__ SAMP) {     // [t, 256]
  const int lane = threadIdx.x & 31;
  const int wave = threadIdx.x >> 5;
  int gw = blockIdx.x * 8 + wave;  // t*NH + h
  int t = gw >> 3;
  int h = gw & 7;
  int b = t & (BB - 1);
  const int HHs[NLV] = {100, 50, 25, 13};
  const int WWs[NLV] = {100, 50, 25, 13};
  const int LST[NLV] = {0, 10000, 12500, 13125};
  float out = 0.f;
#pragma unroll
  for (int l = 0; l < NLV; ++l) {
    const int hh = HHs[l], ww = WWs[l];
    float rx = REF[((size_t)t * NLV + l) * 2 + 0];
    float ry = REF[((size_t)t * NLV + l) * 2 + 1];
#pragma unroll
    for (int p = 0; p < NPT; ++p) {
      size_t oidx = (size_t)t * 256 + (size_t)(((h * NLV + l) * NPT + p) * 2);
      float ox = OFF[oidx], oy = OFF[oidx + 1];
      float x = rx * (float)ww + ox - 0.5f;
      float y = ry * (float)hh + oy - 0.5f;
      float x0f = floorf(x), y0f = floorf(y);
      float wx = x - x0f, wy = y - y0f;
      int x0 = (int)x0f, y0 = (int)y0f;
      float a = AW[(size_t)t * 128 + h * 16 + l * NPT + p];
      float sval = 0.f;
#pragma unroll
      for (int cy = 0; cy < 2; ++cy) {
#pragma unroll
        for (int cx = 0; cx < 2; ++cx) {
          int xi = x0 + cx, yi = y0 + cy;
          bool valid = (xi >= 0) && (xi < ww) && (yi >= 0) && (yi < hh);
          int xc = xi < 0 ? 0 : (xi > ww - 1 ? ww - 1 : xi);
          int yc = yi < 0 ? 0 : (yi > hh - 1 ? hh - 1 : yi);
          int lin = LST[l] + yc * ww + xc;
          float vv = VAL[((size_t)lin * BB + b) * 256 + h * HD + lane];
          float wgt = (cx ? wx : 1.f - wx) * (cy ? wy : 1.f - wy);
          sval += valid ? wgt * vv : 0.f;
        }
      }
      out = fmaf(a, sval, out);
    }
  }
  SAMP[(size_t)t * 256 + h * HD + lane] = out;
}

// ---------------------------------------------------------------------------
extern "C" void kernel_launch(void* const* d_in, const int* in_sizes, int n_in,
                              void* d_out, int out_size, void* d_ws,
                              size_t ws_size, hipStream_t stream) {
  (void)in_sizes; (void)n_in; (void)out_size; (void)ws_size;
  const float* tgt  = (const float*)d_in[0];
  const float* pos  = (const float*)d_in[1];
  const float* ref  = (const float*)d_in[2];
  const float* mem  = (const float*)d_in[3];
  const float* in_w = (const float*)d_in[4];
  const float* in_b = (const float*)d_in[5];
  const float* op_w = (const float*)d_in[6];
  const float* op_b = (const float*)d_in[7];
  const float* so_w = (const float*)d_in[8];
  const float* so_b = (const float*)d_in[9];
  const float* aw_w = (const float*)d_in[10];
  const float* aw_b = (const float*)d_in[11];
  const float* vp_w = (const float*)d_in[12];
  const float* vp_b = (const float*)d_in[13];
  const float* mo_w = (const float*)d_in[14];
  const float* mo_b = (const float*)d_in[15];
  const float* l1_w = (const float*)d_in[16];
  const float* l1_b = (const float*)d_in[17];
  const float* l2_w = (const float*)d_in[18];
  const float* l2_b = (const float*)d_in[19];
  const float* n1g = (const float*)d_in[20];
  const float* n1b = (const float*)d_in[21];
  const float* n2g = (const float*)d_in[22];
  const float* n2b = (const float*)d_in[23];
  const float* n3g = (const float*)d_in[24];
  const float* n3b = (const float*)d_in[25];

  float* ws = (float*)d_ws;
  const size_t N1 = (size_t)M1 * DIM;  // 3,686,400 floats per token slot
  float* X   = ws;                     // x / sa / q2 / t3
  float* QKb = ws + N1;                // QK (2 slots) / off / ffn-out
  float* AWR = ws + 2 * N1;            // raw attn weights [t,128]
  float* AW  = ws + 2 * N1 + N1 / 2;   // softmaxed attn weights
  float* Vb  = ws + 3 * N1;            // self-attn V / deform samp
  float* Ob  = ws + 4 * N1;            // attn out / cross-attn out
  float* T2  = ws + 5 * N1;            // tgt after norm2
  float* VAL = ws + 6 * N1;            // value proj (54.4M f) / ffn hidden

  dim3 blk(256);
  int n4 = (int)(N1 / 4);

  // x = tgt + pos
  add_kernel<<<(n4 + 255) / 256, blk, 0, stream>>>(tgt, pos, X, n4);
  // Q,K = x @ Wqk^T + bqk   (N=512)
  gemm_kernel<<<dim3(8, M1 / 64), blk, 0, stream>>>(X, in_w, in_b, QKb, M1, 512, DIM, 0);
  // V = tgt @ Wv^T + bv
  gemm_kernel<<<dim3(4, M1 / 64), blk, 0, stream>>>(tgt, in_w + 512 * DIM, in_b + 512, Vb, M1, 256, DIM, 0);
  // flash self-attention
  flash_kernel<<<BB * NH * 29, blk, 0, stream>>>(QKb, Vb, Ob);
  // sa = o @ Wout^T + bout
  gemm_kernel<<<dim3(4, M1 / 64), blk, 0, stream>>>(Ob, op_w, op_b, X, M1, 256, DIM, 0);
  // tgt2 = LN(tgt + sa)  [norm2]
  ln_kernel<<<M1 / 8, blk, 0, stream>>>(tgt, X, n2g, n2b, T2, M1);
  // q2 = tgt2 + pos
  add_kernel<<<(n4 + 255) / 256, blk, 0, stream>>>(T2, pos, X, n4);
  // sampling offsets
  gemm_kernel<<<dim3(4, M1 / 64), blk, 0, stream>>>(X, so_w, so_b, QKb, M1, 256, DIM, 0);
  // attention weights (raw) then softmax over 16
  gemm_kernel<<<dim3(2, M1 / 64), blk, 0, stream>>>(X, aw_w, aw_b, AWR, M1, 128, DIM, 0);
  softmax16_kernel<<<(M1 * NH + 255) / 256, blk, 0, stream>>>(AWR, AW, M1 * NH);
  // value projection over full memory (M = 212704, tail-guarded)
  gemm_kernel<<<dim3(4, (MV + 63) / 64), blk, 0, stream>>>(mem, vp_w, vp_b, VAL, MV, 256, DIM, 0);
  // deformable bilinear sampling + weighting
  deform_kernel<<<(M1 * NH) / 8, blk, 0, stream>>>(ref, QKb, AW, VAL, Vb);
  // ca = samp @ Wms^T + bms
  gemm_kernel<<<dim3(4, M1 / 64), blk, 0, stream>>>(Vb, mo_w, mo_b, Ob, M1, 256, DIM, 0);
  // tgt3 = LN(tgt2 + ca)  [norm1]
  ln_kernel<<<M1 / 8, blk, 0, stream>>>(T2, Ob, n1g, n1b, X, M1);
  // FFN: H = relu(tgt3 @ W1^T + b1)
  gemm_kernel<<<dim3(DFF / 64, M1 / 64), blk, 0, stream>>>(X, l1_w, l1_b, VAL, M1, DFF, DIM, 1);
  // t2f = H @ W2^T + b2
  gemm_kernel<<<dim3(4, M1 / 64), blk, 0, stream>>>(VAL, l2_w, l2_b, QKb, M1, 256, DFF, 0);
  // out = LN(tgt3 + t2f)  [norm3]
  ln_kernel<<<M1 / 8, blk, 0, stream>>>(X, QKb, n3g, n3b, (float*)d_out, M1);
}